// GraphEncoder_55190329753873
// MI455X (gfx1250) — compile-verified
//
#include <hip/hip_runtime.h>

typedef __attribute__((ext_vector_type(2))) float v2f;
typedef __attribute__((ext_vector_type(8))) float v8f;

#define EPS 1e-5f

__device__ __forceinline__ v8f wmma_f32(v2f a, v2f b, v8f c) {
    // D = A(16x4, f32) * B(4x16, f32) + C(16x16, f32)
    return __builtin_amdgcn_wmma_f32_16x16x4_f32(
        /*neg_a=*/false, a, /*neg_b=*/false, b,
        /*c_mod=*/(short)0, c, /*reuse_a=*/false, /*reuse_b=*/false);
}

// ---------------------------------------------------------------- utilities
__global__ void zero_f32(float* __restrict__ p, long long count) {
    long long i = (long long)blockIdx.x * blockDim.x + threadIdx.x;
    long long stride = (long long)gridDim.x * blockDim.x;
    for (; i < count; i += stride) p[i] = 0.0f;
}

__global__ void deg_kernel(const int* __restrict__ dst, float* __restrict__ deg, int E) {
    int i = blockIdx.x * blockDim.x + threadIdx.x;
    int stride = gridDim.x * blockDim.x;
    for (; i < E; i += stride) atomicAdd(&deg[dst[i]], 1.0f);
}

// ------------------------------------------------------------- edge scatter
// One wave32 per edge: lane covers 4 of 128 h-floats and 1 of 32 rel-floats.
// hsum[dst] += h[src]; relsum[dst] += rel_emb_l[edge_rel]
__global__ __launch_bounds__(256) void scatter_kernel(
    const float* __restrict__ h, const float* __restrict__ rel_table,
    const int* __restrict__ src, const int* __restrict__ dst,
    const int* __restrict__ erel, float* __restrict__ hsum,
    float* __restrict__ relsum, int E) {
    const int lane = threadIdx.x & 31;
    long long w  = ((long long)blockIdx.x * blockDim.x + threadIdx.x) >> 5;
    long long nw = ((long long)gridDim.x * blockDim.x) >> 5;
    for (long long e = w; e < E; e += nw) {
        const int s = src[e], d = dst[e], r = erel[e];
        const float4 v = *(const float4*)(h + (long long)s * 128 + lane * 4);
        float* hd = hsum + (long long)d * 128 + lane * 4;
        atomicAdd(hd + 0, v.x);
        atomicAdd(hd + 1, v.y);
        atomicAdd(hd + 2, v.z);
        atomicAdd(hd + 3, v.w);
        atomicAdd(relsum + (long long)d * 32 + lane,
                  rel_table[(long long)r * 32 + lane]);
    }
}

// --------------------------------------------------- input projection (WMMA)
// h0 = relu(concat(x[16x128], label_emb[label][16x32]) @ W[160x128] + b)
__global__ __launch_bounds__(256) void inproj_kernel(
    const float* __restrict__ x, const int* __restrict__ label,
    const float* __restrict__ label_emb, const float* __restrict__ W,
    const float* __restrict__ bias, float* __restrict__ hout, int n) {
    const int tid = threadIdx.x, lane = tid & 31, wave = tid >> 5;
    const int hi = lane >> 4;       // upper half-wave handles K+2,K+3 / M+8
    const int lrow = lane & 15;     // A: M row, B/C/D: N column within tile
    const int col = wave * 16 + lrow;

    __shared__ float A[16 * 164];   // 16 x 160, padded stride

    // Preload B fragments (one 160x16 weight column slab per wave, in VGPRs)
    v2f B[40];
#pragma unroll
    for (int k = 0; k < 40; ++k) {
        const int kk = k * 4 + hi * 2;
        B[k].x = W[(kk + 0) * 128 + col];
        B[k].y = W[(kk + 1) * 128 + col];
    }
    const float bcol = bias[col];

    const int ntiles = (n + 15) >> 4;
    for (int t = blockIdx.x; t < ntiles; t += gridDim.x) {
        const int base = t * 16;
        for (int idx = tid; idx < 16 * 128; idx += 256) {
            const int r = idx >> 7, c = idx & 127;
            const int node = base + r;
            A[r * 164 + c] = (node < n) ? x[(long long)node * 128 + c] : 0.0f;
        }
        for (int idx = tid; idx < 16 * 32; idx += 256) {
            const int r = idx >> 5, c = idx & 31;
            const int node = base + r;
            float v = 0.0f;
            if (node < n) v = label_emb[(long long)label[node] * 32 + c];
            A[r * 164 + 128 + c] = v;
        }
        __syncthreads();

        v8f acc = {};
#pragma unroll
        for (int k = 0; k < 40; ++k) {
            const int kk = k * 4 + hi * 2;
            v2f a;
            a.x = A[lrow * 164 + kk + 0];
            a.y = A[lrow * 164 + kk + 1];
            acc = wmma_f32(a, B[k], acc);
        }
#pragma unroll
        for (int i = 0; i < 8; ++i) {
            const int node = base + i + hi * 8;
            if (node < n) {
                const float v = acc[i] + bcol;
                hout[(long long)node * 128 + col] = v > 0.0f ? v : 0.0f;
            }
        }
        __syncthreads();
    }
}

// --------------------------------------------- fused node update (WMMA + LN)
// out = h@Ws + b + inv_deg * (hsum@Wn + relsum@Wr);  h' = LN(relu(out))
__global__ __launch_bounds__(256) void update_kernel(
    const float* __restrict__ hin, float* __restrict__ hout,
    const float* __restrict__ hsum, const float* __restrict__ relsum,
    const float* __restrict__ deg,
    const float* __restrict__ Wself, const float* __restrict__ Wneigh,
    const float* __restrict__ Wrel, const float* __restrict__ bias,
    const float* __restrict__ ln_g, const float* __restrict__ ln_b, int n) {
    const int tid = threadIdx.x, lane = tid & 31, wave = tid >> 5;
    const int hi = lane >> 4, lrow = lane & 15;
    const int col = wave * 16 + lrow;

    __shared__ float Ah[16 * 132];
    __shared__ float As[16 * 132];
    __shared__ float Ar[16 * 36];
    __shared__ float Out[16 * 132];
    __shared__ float invd[16];

    // Register-resident weight column slabs (loaded once per block)
    v2f Bs[32], Bn[32], Br[8];
#pragma unroll
    for (int k = 0; k < 32; ++k) {
        const int kk = k * 4 + hi * 2;
        Bs[k].x = Wself[(kk + 0) * 128 + col];
        Bs[k].y = Wself[(kk + 1) * 128 + col];
        Bn[k].x = Wneigh[(kk + 0) * 128 + col];
        Bn[k].y = Wneigh[(kk + 1) * 128 + col];
    }
#pragma unroll
    for (int k = 0; k < 8; ++k) {
        const int kk = k * 4 + hi * 2;
        Br[k].x = Wrel[(kk + 0) * 128 + col];
        Br[k].y = Wrel[(kk + 1) * 128 + col];
    }
    const float bcol = bias[col];

    const int ntiles = (n + 15) >> 4;
    for (int t = blockIdx.x; t < ntiles; t += gridDim.x) {
        const int base = t * 16;
        for (int idx = tid; idx < 16 * 128; idx += 256) {
            const int r = idx >> 7, c = idx & 127;
            const long long node = base + r;
            const bool ok = node < n;
            Ah[r * 132 + c] = ok ? hin[node * 128 + c] : 0.0f;
            As[r * 132 + c] = ok ? hsum[node * 128 + c] : 0.0f;
        }
        for (int idx = tid; idx < 16 * 32; idx += 256) {
            const int r = idx >> 5, c = idx & 31;
            const long long node = base + r;
            Ar[r * 36 + c] = (node < n) ? relsum[node * 32 + c] : 0.0f;
        }
        if (tid < 16) {
            const int node = base + tid;
            const float dg = (node < n) ? deg[node] : 0.0f;
            invd[tid] = dg > 0.0f ? 1.0f / dg : 0.0f;
        }
        __syncthreads();

        v8f acc_s = {};   // h @ Wself
        v8f acc_a = {};   // hsum @ Wneigh + relsum @ Wrel (scaled by inv_deg)
#pragma unroll
        for (int k = 0; k < 32; ++k) {
            const int kk = k * 4 + hi * 2;
            v2f ah, as;
            ah.x = Ah[lrow * 132 + kk];  ah.y = Ah[lrow * 132 + kk + 1];
            as.x = As[lrow * 132 + kk];  as.y = As[lrow * 132 + kk + 1];
            acc_s = wmma_f32(ah, Bs[k], acc_s);
            acc_a = wmma_f32(as, Bn[k], acc_a);
        }
#pragma unroll
        for (int k = 0; k < 8; ++k) {
            const int kk = k * 4 + hi * 2;
            v2f ar;
            ar.x = Ar[lrow * 36 + kk];  ar.y = Ar[lrow * 36 + kk + 1];
            acc_a = wmma_f32(ar, Br[k], acc_a);
        }
#pragma unroll
        for (int i = 0; i < 8; ++i) {
            const int row = i + hi * 8;
            const float v = acc_s[i] + bcol + invd[row] * acc_a[i];
            Out[row * 132 + col] = v > 0.0f ? v : 0.0f;   // relu before LN
        }
        __syncthreads();

        // LayerNorm: wave w reduces rows 2w, 2w+1 (128 cols = 4 per lane)
#pragma unroll
        for (int rr = 0; rr < 2; ++rr) {
            const int row = wave * 2 + rr;
            const int node = base + row;
            const float v0 = Out[row * 132 + lane * 4 + 0];
            const float v1 = Out[row * 132 + lane * 4 + 1];
            const float v2 = Out[row * 132 + lane * 4 + 2];
            const float v3 = Out[row * 132 + lane * 4 + 3];
            float s = v0 + v1 + v2 + v3;
            float q = v0 * v0 + v1 * v1 + v2 * v2 + v3 * v3;
#pragma unroll
            for (int off = 16; off >= 1; off >>= 1) {
                s += __shfl_xor(s, off, 32);
                q += __shfl_xor(q, off, 32);
            }
            const float mu = s * (1.0f / 128.0f);
            const float var = q * (1.0f / 128.0f) - mu * mu;
            const float rstd = rsqrtf(var + EPS);
            if (node < n) {
                float* o = hout + (long long)node * 128 + lane * 4;
                const int c = lane * 4;
                o[0] = (v0 - mu) * rstd * ln_g[c + 0] + ln_b[c + 0];
                o[1] = (v1 - mu) * rstd * ln_g[c + 1] + ln_b[c + 1];
                o[2] = (v2 - mu) * rstd * ln_g[c + 2] + ln_b[c + 2];
                o[3] = (v3 - mu) * rstd * ln_g[c + 3] + ln_b[c + 3];
            }
        }
        __syncthreads();
    }
}

// ------------------------------------------------------------------- launch
extern "C" void kernel_launch(void* const* d_in, const int* in_sizes, int n_in,
                              void* d_out, int out_size, void* d_ws, size_t ws_size,
                              hipStream_t stream) {
    const float* x          = (const float*)d_in[0];
    const int*   label      = (const int*)d_in[1];
    const int*   edge_index = (const int*)d_in[2];
    const int*   edge_rel   = (const int*)d_in[3];
    const float* label_emb  = (const float*)d_in[4];
    const float* in_proj_w  = (const float*)d_in[5];
    const float* in_proj_b  = (const float*)d_in[6];
    const float* rel_emb    = (const float*)d_in[7];
    const float* lin_neigh_w = (const float*)d_in[8];
    const float* lin_self_w  = (const float*)d_in[9];
    const float* lin_self_b  = (const float*)d_in[10];
    const float* lin_rel_w   = (const float*)d_in[11];
    const float* ln_g        = (const float*)d_in[12];
    const float* ln_b        = (const float*)d_in[13];

    const int n = in_sizes[1];       // N nodes
    const int E = in_sizes[3];       // edges
    const int* src = edge_index;
    const int* dst = edge_index + E;

    float* buf0 = (float*)d_ws;                 // h after input proj   [n,128]
    float* hsum = buf0 + (size_t)n * 128;       // segment_sum(h[src])  [n,128]
    float* rels = hsum + (size_t)n * 128;       // segment_sum(rel)     [n,32]
    float* degp = rels + (size_t)n * 32;        // in-degree            [n]
    float* hout = (float*)d_out;

    // degrees (once)
    zero_f32<<<256, 256, 0, stream>>>(degp, n);
    deg_kernel<<<1024, 256, 0, stream>>>(dst, degp, E);

    // input projection -> buf0
    inproj_kernel<<<512, 256, 0, stream>>>(x, label, label_emb, in_proj_w,
                                           in_proj_b, buf0, n);

    const float* h_l = buf0;
    for (int l = 0; l < 2; ++l) {
        // hsum and rels are contiguous: zero both in one pass (n*160 floats)
        zero_f32<<<2048, 256, 0, stream>>>(hsum, (long long)n * 160);
        scatter_kernel<<<4096, 256, 0, stream>>>(
            h_l, rel_emb + (size_t)l * 1024 * 32, src, dst, edge_rel,
            hsum, rels, E);
        update_kernel<<<512, 256, 0, stream>>>(
            h_l, hout, hsum, rels, degp,
            lin_self_w + (size_t)l * 128 * 128,
            lin_neigh_w + (size_t)l * 128 * 128,
            lin_rel_w + (size_t)l * 32 * 128,
            lin_self_b + (size_t)l * 128,
            ln_g + (size_t)l * 128, ln_b + (size_t)l * 128, n);
        h_l = hout;   // layer 1 reads layer-0 output from d_out
    }
}